// SPMotifNet_10007273799859
// MI455X (gfx1250) — compile-verified
//
#include <hip/hip_runtime.h>
#include <hip/hip_bf16.h>

typedef __attribute__((ext_vector_type(2))) float v2f;
typedef __attribute__((ext_vector_type(8))) float v8f;

#define H 64
#define FIN 4
#define CCLS 3
#define LNUM 2

// ---------------- zero workspace tail (wsum/pool/cnt) ----------------
__global__ void zero_kernel(float* __restrict__ p, int n) {
    int i = blockIdx.x * blockDim.x + threadIdx.x;
    if (i < n) p[i] = 0.0f;
}

// ---------------- h = x @ w_emb + b_emb  ([N,4]x[4,64]) ----------------
__global__ void emb_kernel(const float* __restrict__ x,
                           const float* __restrict__ w,
                           const float* __restrict__ b,
                           float* __restrict__ h, int total) {
    int i = blockIdx.x * blockDim.x + threadIdx.x;
    if (i >= total) return;
    int n = i >> 6, f = i & 63;
    const float* xp = x + n * FIN;
    float s = b[f];
#pragma unroll
    for (int j = 0; j < FIN; ++j) s += xp[j] * w[j * H + f];
    h[i] = s;
}

// ---------------- wsum[i] = sum of edge_attr over incoming edges ----------------
__global__ void wsum_kernel(const int* __restrict__ dst,
                            const float* __restrict__ ea,
                            float* __restrict__ wsum, int E) {
    int e = blockIdx.x * blockDim.x + threadIdx.x;
    if (e < E) atomicAdd(&wsum[dst[e]], ea[e]);
}

// ---------------- fused triple GEMM via V_WMMA_F32_16X16X4_F32 ----------------
// Per wave: one 16x16 output tile (rows m0..m0+15, cols n0..n0+15) for each of
// A1 = h@W1 + b1, B2 = h@W2, C3 = h@W3 + b3.  4 waves/block cover all 64 cols.
//
// Layouts (ISA 7.12.2, wave32):
//   A (16x4 f32):  lane l: m = l&15, k = 2*(l>>4) + {vgpr0,vgpr1}
//   B (4x16 f32):  lane l: n = l&15, k = 2*(l>>4) + {vgpr0,vgpr1}
//   C/D (16x16):   lane l: n = l&15, vgpr v: m = v + 8*(l>>4)
__global__ __launch_bounds__(128) void gemm3_kernel(
    const float* __restrict__ h,
    const float* __restrict__ W1, const float* __restrict__ b1,
    const float* __restrict__ W2,
    const float* __restrict__ W3, const float* __restrict__ b3,
    float* __restrict__ outA, float* __restrict__ outB, float* __restrict__ outC) {
    const int lane = threadIdx.x & 31;
    const int wave = threadIdx.x >> 5;       // 0..3 -> column tile
    const int m0   = blockIdx.x * 16;
    const int n0   = wave * 16;
    const int mr   = lane & 15;
    const int half = lane >> 4;              // 0 or 1
    const int col  = n0 + mr;

    v8f accA = {}, accB = {}, accC = {};
#pragma unroll
    for (int k = 0; k < H; k += 4) {
        const int ka = k + (half << 1);
        v2f af, b1f, b2f, b3f;
        const float* hp = h + (size_t)(m0 + mr) * H + ka;
        af.x = hp[0];
        af.y = hp[1];
        b1f.x = W1[(size_t)ka * H + col];       b1f.y = W1[(size_t)(ka + 1) * H + col];
        b2f.x = W2[(size_t)ka * H + col];       b2f.y = W2[(size_t)(ka + 1) * H + col];
        b3f.x = W3[(size_t)ka * H + col];       b3f.y = W3[(size_t)(ka + 1) * H + col];
        accA = __builtin_amdgcn_wmma_f32_16x16x4_f32(false, af, false, b1f, (short)0, accA, false, false);
        accB = __builtin_amdgcn_wmma_f32_16x16x4_f32(false, af, false, b2f, (short)0, accB, false, false);
        accC = __builtin_amdgcn_wmma_f32_16x16x4_f32(false, af, false, b3f, (short)0, accC, false, false);
    }

    const float biasA = b1[col];
    const float biasC = b3[col];
#pragma unroll
    for (int v = 0; v < 8; ++v) {
        const int row = m0 + v + half * 8;
        const size_t o = (size_t)row * H + col;
        outA[o] = accA[v] + biasA;
        outB[o] = accB[v];
        outC[o] = accC[v] + biasC;
    }
}

// scalar tail for rows not covered by 16-row tiles (N % 16 != 0)
__global__ void gemm3_tail_kernel(
    const float* __restrict__ h,
    const float* __restrict__ W1, const float* __restrict__ b1,
    const float* __restrict__ W2,
    const float* __restrict__ W3, const float* __restrict__ b3,
    float* __restrict__ outA, float* __restrict__ outB, float* __restrict__ outC,
    int rowStart, int N) {
    int i = blockIdx.x * blockDim.x + threadIdx.x;
    int rows = N - rowStart;
    if (i >= rows * H) return;
    int n = rowStart + i / H;
    int f = i % H;
    float sa = b1[f], sb = 0.0f, sc = b3[f];
    const float* hp = h + (size_t)n * H;
    for (int k = 0; k < H; ++k) {
        float hv = hp[k];
        sa += hv * W1[k * H + f];
        sb += hv * W2[k * H + f];
        sc += hv * W3[k * H + f];
    }
    size_t o = (size_t)n * H + f;
    outA[o] = sa; outB[o] = sb; outC[o] = sc;
}

// ---------------- edge scatter: C[dst] += a[src] * w  (a-term only) ----------------
// thread handles one edge x 4 consecutive features (float4 gather, 4 atomics)
__global__ void edge_kernel(const int* __restrict__ src, const int* __restrict__ dst,
                            const float* __restrict__ ea,
                            const float* __restrict__ A, float* __restrict__ C,
                            int total) {
    int idx = blockIdx.x * blockDim.x + threadIdx.x;
    if (idx >= total) return;
    int e = idx >> 4;
    int f = (idx & 15) << 2;
    int s = src[e], d = dst[e];
    float w = ea[e];
    const float4 av = *(const float4*)(A + (size_t)s * H + f);
    float* cp = C + (size_t)d * H + f;
    atomicAdd(cp + 0, av.x * w);
    atomicAdd(cp + 1, av.y * w);
    atomicAdd(cp + 2, av.z * w);
    atomicAdd(cp + 3, av.w * w);
}

// ---------------- h = relu(C - B*wsum)  (applies -b[i]*sum(w) term) ----------------
__global__ void relu_kernel(const float* __restrict__ C, const float* __restrict__ B,
                            const float* __restrict__ wsum, float* __restrict__ h,
                            int total) {
    int i = blockIdx.x * blockDim.x + threadIdx.x;
    if (i >= total) return;
    int n = i >> 6;
    h[i] = fmaxf(C[i] - B[i] * wsum[n], 0.0f);
}

// ---------------- global mean pool (sums + counts) ----------------
__global__ void pool_kernel(const float* __restrict__ h, const int* __restrict__ batch,
                            float* __restrict__ pool, float* __restrict__ cnt,
                            int total) {
    int i = blockIdx.x * blockDim.x + threadIdx.x;
    if (i >= total) return;
    int n = i >> 6, f = i & 63;
    int g = batch[n];
    atomicAdd(&pool[(size_t)g * H + f], h[i]);
    if (f == 0) atomicAdd(&cnt[g], 1.0f);
}

// ---------------- MLP head: relu(gx@W1+b1)@W2+b2, one block per graph ----------------
__global__ __launch_bounds__(128) void head_kernel(
    const float* __restrict__ pool, const float* __restrict__ cnt,
    const float* __restrict__ w1, const float* __restrict__ b1,
    const float* __restrict__ w2, const float* __restrict__ b2,
    float* __restrict__ out) {
    __shared__ float gx[H];
    __shared__ float hid[2 * H];
    int g = blockIdx.x;
    int t = threadIdx.x;
    if (t < H) {
        float c = fmaxf(cnt[g], 1.0f);
        gx[t] = pool[(size_t)g * H + t] / c;
    }
    __syncthreads();
    float s = b1[t];
    for (int f = 0; f < H; ++f) s += gx[f] * w1[f * 2 * H + t];
    hid[t] = fmaxf(s, 0.0f);
    __syncthreads();
    if (t < CCLS) {
        float o = b2[t];
        for (int j = 0; j < 2 * H; ++j) o += hid[j] * w2[j * CCLS + t];
        out[(size_t)g * CCLS + t] = o;
    }
}

extern "C" void kernel_launch(void* const* d_in, const int* in_sizes, int n_in,
                              void* d_out, int out_size, void* d_ws, size_t ws_size,
                              hipStream_t stream) {
    const float* x      = (const float*)d_in[0];
    const int*   eidx   = (const int*)  d_in[1];
    const float* ea     = (const float*)d_in[2];
    const int*   batch  = (const int*)  d_in[3];
    const float* w_emb  = (const float*)d_in[4];
    const float* b_emb  = (const float*)d_in[5];
    const float* lin1_w = (const float*)d_in[6];
    const float* lin1_b = (const float*)d_in[7];
    const float* lin2_w = (const float*)d_in[8];
    const float* lin3_w = (const float*)d_in[9];
    const float* lin3_b = (const float*)d_in[10];
    const float* mlp_w1 = (const float*)d_in[11];
    const float* mlp_b1 = (const float*)d_in[12];
    const float* mlp_w2 = (const float*)d_in[13];
    const float* mlp_b2 = (const float*)d_in[14];
    float* out = (float*)d_out;

    const int N = in_sizes[3];          // batch has N entries
    const int E = in_sizes[2];          // edge_attr has E entries
    const int G = out_size / CCLS;
    const int* src = eidx;
    const int* dst = eidx + E;

    // workspace layout: h | bufA | bufB | bufC | wsum | pool | cnt
    float* ws   = (float*)d_ws;
    float* h    = ws;
    float* bufA = h    + (size_t)N * H;
    float* bufB = bufA + (size_t)N * H;
    float* bufC = bufB + (size_t)N * H;
    float* wsum = bufC + (size_t)N * H;
    float* pool = wsum + N;
    float* cnt  = pool + (size_t)G * H;

    const int zcount = N + G * H + G;   // wsum..cnt contiguous
    zero_kernel<<<(zcount + 255) / 256, 256, 0, stream>>>(wsum, zcount);

    emb_kernel<<<(N * H + 255) / 256, 256, 0, stream>>>(x, w_emb, b_emb, h, N * H);
    wsum_kernel<<<(E + 255) / 256, 256, 0, stream>>>(dst, ea, wsum, E);

    const int tiles  = N / 16;
    const int tail   = tiles * 16;
    const int etotal = E * 16;
    for (int l = 0; l < LNUM; ++l) {
        const float* W1 = lin1_w + (size_t)l * H * H;
        const float* B1 = lin1_b + (size_t)l * H;
        const float* W2 = lin2_w + (size_t)l * H * H;
        const float* W3 = lin3_w + (size_t)l * H * H;
        const float* B3 = lin3_b + (size_t)l * H;
        if (tiles > 0)
            gemm3_kernel<<<tiles, 128, 0, stream>>>(h, W1, B1, W2, W3, B3,
                                                    bufA, bufB, bufC);
        if (tail < N) {
            int rem = (N - tail) * H;
            gemm3_tail_kernel<<<(rem + 255) / 256, 256, 0, stream>>>(
                h, W1, B1, W2, W3, B3, bufA, bufB, bufC, tail, N);
        }
        edge_kernel<<<(etotal + 255) / 256, 256, 0, stream>>>(src, dst, ea, bufA, bufC, etotal);
        relu_kernel<<<(N * H + 255) / 256, 256, 0, stream>>>(bufC, bufB, wsum, h, N * H);
    }

    pool_kernel<<<(N * H + 255) / 256, 256, 0, stream>>>(h, batch, pool, cnt, N * H);
    head_kernel<<<G, 128, 0, stream>>>(pool, cnt, mlp_w1, mlp_b1, mlp_w2, mlp_b2, out);
}